// VonMisesFisher_36197984370726
// MI455X (gfx1250) — compile-verified
//
#include <hip/hip_runtime.h>
#include <hip/hip_bf16.h>

// ---------------------------------------------------------------------------
// vMF sampler for MI455X (gfx1250, wave32).
//   - 16 rows per wave; 3 row-reductions done with v_wmma_f32_16x16x4_f32
//     (A=ones, B=lane partials -> C = row sums broadcast per lane).
//   - loc staged via global_load_async_to_lds_b128, double buffered,
//     synchronized with s_wait_asynccnt.
//   - Raw HW transcendentals (v_sqrt/v_rcp/v_rsq/v_log/v_sin/v_cos).
//   - Cheap 32-bit counter-based RNG (lowbias32) instead of 64-bit splitmix.
// ---------------------------------------------------------------------------

#define M        512
#define ROWS_PW  16          // rows per wave (fixed by WMMA 16x16 tile)
#define WAVES    8           // waves per block
#define ROWS_PB  (ROWS_PW * WAVES)   // 128 rows per block
#define TILE     32          // columns per LDS tile
#define NTILES   (M / TILE)  // 16
#define CHUNKS   (TILE / 4)  // 8 K-chunks of 4 columns
#define PADS     36          // padded LDS row stride (zero bank conflicts)

typedef float v2f __attribute__((ext_vector_type(2)));
typedef float v8f __attribute__((ext_vector_type(8)));

// Raw hardware transcendentals (single-instruction lowering).
__device__ __forceinline__ float hw_sqrt(float x) { return __builtin_amdgcn_sqrtf(x); }
__device__ __forceinline__ float hw_rcp(float x)  { return __builtin_amdgcn_rcpf(x); }
__device__ __forceinline__ float hw_rsq(float x)  { return __builtin_amdgcn_rsqf(x); }

// lowbias32: high-quality 32-bit mixer, ~8 VALU ops.
__device__ __forceinline__ unsigned hash32(unsigned x) {
    x ^= x >> 16; x *= 0x7FEB352Du;
    x ^= x >> 15; x *= 0x846CA68Bu;
    x ^= x >> 16;
    return x;
}
__device__ __forceinline__ float u24(unsigned h) {        // (0,1)
    return (float)(h >> 8) * 0x1p-24f + 0x1p-25f;
}

// Box-Muller pair for tangential vector v: deterministic in (row, col-pair j).
__device__ __forceinline__ float2 vmf_normal_pair(unsigned row, unsigned j) {
    unsigned s0 = (row << 9) ^ j;
    float u1 = u24(hash32(s0 ^ 0x9E3779B9u));
    float u2 = u24(hash32(s0 ^ 0x85EBCA6Bu));
    float rr = hw_sqrt(-2.0f * __logf(u1));
    float sn, cs;
    __sincosf(6.28318530717958647f * u2, &sn, &cs);
    return make_float2(rr * cs, rr * sn);
}

// Wood's rejection sampler for w = <z, mu>; Beta(255.5,255.5) ~ N(0.5, 1/(4*512)).
__device__ __forceinline__ float sample_w(float kappa, unsigned row) {
    const float m1 = 511.0f;
    float c  = hw_sqrt(4.0f * kappa * kappa + m1 * m1);
    float bt = (c - 2.0f * kappa) * (1.0f / 511.0f);
    float ba = 127.75f * hw_rcp(kappa);               // (m-1)/(4k)
    float s  = fminf(fmaxf(kappa - 10.0f, 0.0f), 1.0f);
    float b  = ba * s + bt * (1.0f - s);
    float a  = (m1 + 2.0f * kappa + c) * 0.25f;
    float ab4 = 4.0f * a * b;
    float d  = ab4 * hw_rcp(1.0f + b) - m1 * __logf(m1);

    float w = 0.0f;
    bool active = true;
#pragma unroll 1
    for (int it = 0; it < 50; ++it) {
        unsigned s0 = (row << 6) ^ (unsigned)it;
        // one normal draw (Box-Muller, cosine branch)
        float u1 = u24(hash32(s0 ^ 0xA511E9B3u));
        float u2 = u24(hash32(s0 ^ 0x2545F491u));
        float n01 = hw_sqrt(-2.0f * __logf(u1)) *
                    __cosf(6.28318530717958647f * u2);
        float e = 0.5f + 0.011048543456f * n01;       // sigma = 1/(4*sqrt(512))
        e = fminf(fmaxf(e, 1e-6f), 1.0f - 1e-6f);
        float u = u24(hash32(s0 ^ 0x68E31DA4u));

        float rden = hw_rcp(1.0f - (1.0f - b) * e);   // 1/denom
        float w_   = (1.0f - (1.0f + b) * e) * rden;
        float t    = 0.5f * ab4 * rden;               // 2ab/denom
        bool accept = (m1 * __logf(t) - t + d) > __logf(u);
        w      = (active && accept) ? w_ : w;
        active = active && !accept;
        if (!__any((int)active)) break;               // wave-uniform early exit
    }
    return w;
}

__global__ __launch_bounds__(256) void
vmf_kernel(const float* __restrict__ loc, const float* __restrict__ scale,
           float* __restrict__ out, int nrows) {
    // Double-buffered loc tiles: 8 waves x 2 bufs x 16 rows x 36 floats = 36.9 KB
    __shared__ float lds_loc[WAVES][2][ROWS_PW][PADS];

    const int tid  = threadIdx.x;
    const int wv   = tid >> 5;
    const int lane = tid & 31;
    const int h    = lane >> 4;      // wave half (selects K phases {2h, 2h+1})
    const int r    = lane & 15;      // my row within the 16-row group
    const int qrow = lane >> 3;      // 0..3 (async-copy row group)
    const int qsub = lane & 7;       // 0..7 (float4 slot within a row segment)

    const int R = blockIdx.x * ROWS_PB + wv * ROWS_PW;
    if (R >= nrows) return;
    const int row = R + r;

    // Issue async load of loc tile t into buffer `buf` (4 rows / instruction,
    // each instruction moves 4 fully-used 128B segments).
#define LOAD_TILE(t, buf)                                                      \
    {                                                                          \
        _Pragma("unroll") for (int i_ = 0; i_ < 4; ++i_) {                     \
            int rr_ = i_ * 4 + qrow;                                           \
            unsigned goff_ = ((unsigned)(R + rr_) * (unsigned)M +              \
                              (unsigned)((t) * TILE) + (unsigned)(qsub * 4)) * \
                             4u;                                               \
            unsigned loff_ =                                                   \
                (unsigned)(size_t)&lds_loc[wv][(buf)][rr_][qsub * 4];          \
            asm volatile("global_load_async_to_lds_b128 %0, %1, %2"            \
                         :: "v"(loff_), "v"(goff_), "s"(loc)                   \
                         : "memory");                                          \
        }                                                                      \
    }
#define WAIT_ASYNC_LE1() asm volatile("s_wait_asynccnt 1" ::: "memory")
#define WAIT_ASYNC_0()   asm volatile("s_wait_asynccnt 0" ::: "memory")

    // Kick off first tile; overlap the (long) w sampling with it.
    LOAD_TILE(0, 0);
    float kappa = scale[row];
    float loc00 = loc[(size_t)row * M];          // loc[row][0] (L2 resident)
    float w     = sample_w(kappa, (unsigned)row);

    // ---------------- Pass 1: reductions Su, Sv, Svu --------------------
    float su0 = 0.f, su1 = 0.f, sv0 = 0.f, sv1 = 0.f, suv0 = 0.f, suv1 = 0.f;
#pragma unroll 1
    for (int t = 0; t < NTILES; ++t) {
        if (t + 1 < NTILES) { LOAD_TILE(t + 1, (t + 1) & 1); WAIT_ASYNC_LE1(); }
        else                { WAIT_ASYNC_0(); }
        const float* tl = &lds_loc[wv][t & 1][r][0];
#pragma unroll
        for (int cc = 0; cc < CHUNKS; ++cc) {
            int colL = cc * 4 + 2 * h;
            int col  = t * TILE + colL;          // even global column
            float l0 = tl[colL], l1 = tl[colL + 1];
            float u0 = ((col == 0) ? 1.0f : 0.0f) - l0;
            float u1 = -l1;
            float2 vp = vmf_normal_pair((unsigned)row, (unsigned)(col >> 1));
            float v0 = (col == 0) ? 0.0f : vp.x;  // v has no k=0 component
            float v1 = vp.y;
            su0  += u0 * u0;  su1  += u1 * u1;
            sv0  += v0 * v0;  sv1  += v1 * v1;
            suv0 += v0 * u0;  suv1 += v1 * u1;
        }
    }

    // WMMA reduction: A = ones(16x4), B = lane partials -> C[m,n] = rowsum(n),
    // broadcast across M; lane (l&15)==n reads its own row's sum from c[0].
    v2f onesA = {1.0f, 1.0f};
    v8f zc = {0.f, 0.f, 0.f, 0.f, 0.f, 0.f, 0.f, 0.f};
    v2f bU = {su0, su1}, bV = {sv0, sv1}, bUV = {suv0, suv1};
    v8f cU = __builtin_amdgcn_wmma_f32_16x16x4_f32(false, onesA, false, bU,
                                                   (short)0, zc, false, false);
    v8f cV = __builtin_amdgcn_wmma_f32_16x16x4_f32(false, onesA, false, bV,
                                                   (short)0, zc, false, false);
    v8f cUV = __builtin_amdgcn_wmma_f32_16x16x4_f32(false, onesA, false, bUV,
                                                    (short)0, zc, false, false);
    float Su = cU[0], Sv = cV[0], Suv = cUV[0];

    // Householder coefficients for my row.
    float sf    = hw_sqrt(fmaxf(1.0f - w * w, 1e-10f));
    float xs    = sf * hw_rsq(fmaxf(Sv, 1e-30f));      // sf / ||v||
    float u0raw = 1.0f - loc00;
    float nu    = hw_sqrt(Su) + 1e-5f;                 // ||e1 - loc|| + eps
    float rnu   = hw_rcp(nu);
    float c2    = 2.0f * (w * u0raw + xs * Suv) * rnu * rnu; // applied to raw u

    // ---------------- Pass 2: z = x - c2 * u_raw ------------------------
    LOAD_TILE(0, 0);
#pragma unroll 1
    for (int t = 0; t < NTILES; ++t) {
        if (t + 1 < NTILES) { LOAD_TILE(t + 1, (t + 1) & 1); WAIT_ASYNC_LE1(); }
        else                { WAIT_ASYNC_0(); }
        const float* tl = &lds_loc[wv][t & 1][r][0];
#pragma unroll
        for (int cc = 0; cc < CHUNKS; ++cc) {
            int colL = cc * 4 + 2 * h;
            int col  = t * TILE + colL;
            float l0 = tl[colL], l1 = tl[colL + 1];
            float u0 = ((col == 0) ? 1.0f : 0.0f) - l0;
            float u1 = -l1;
            float2 vp = vmf_normal_pair((unsigned)row, (unsigned)(col >> 1));
            float x0 = (col == 0) ? w : xs * vp.x;
            float x1 = xs * vp.y;
            float z0 = x0 - c2 * u0;
            float z1 = x1 - c2 * u1;
            float2 zz = make_float2(z0, z1);
            *(float2*)&out[(size_t)row * M + col] = zz;  // GL2 merges per line
        }
    }
#undef LOAD_TILE
#undef WAIT_ASYNC_LE1
#undef WAIT_ASYNC_0
}

extern "C" void kernel_launch(void* const* d_in, const int* in_sizes, int n_in,
                              void* d_out, int out_size, void* d_ws, size_t ws_size,
                              hipStream_t stream) {
    const float* loc   = (const float*)d_in[0];
    const float* scale = (const float*)d_in[1];
    float* out = (float*)d_out;
    int nrows = in_sizes[1];                  // B = 32768 (scale is [B,1])
    int blocks = (nrows + ROWS_PB - 1) / ROWS_PB;   // 256 blocks of 256 threads
    vmf_kernel<<<blocks, 256, 0, stream>>>(loc, scale, out, nrows);
}